// PopulationCollage_15204184228038
// MI455X (gfx1250) — compile-verified
//
#include <hip/hip_runtime.h>
#include <hip/hip_bf16.h>
#include <math.h>

// Problem constants (from reference): S,B,C,H,W = 2,100,5,224,224
#define S_   2
#define B_   100
#define C_   5
#define H_   224
#define W_   224
#define HW_  (H_ * W_)
#define WSB_ 112              // B padded to multiple of 16 (7 chunks of 16)
#define REC_ 16               // floats per (s,b) record in ws/LDS (64B, pow2)
#define TILES_X 14            // 14*16 = 224
#define TILES_Y 28            // 28*8  = 224

typedef float v2f __attribute__((ext_vector_type(2)));
typedef float v8f __attribute__((ext_vector_type(8)));

// ---------------------------------------------------------------------------
// Kernel 1: per-(s,b) affine composition (closed form of Msc@Mrot@Mshear@Mtr)
// plus colours, written to d_ws as [S][WSB_][16] floats laid out so the WMMA
// A-fragment is a single unconditional ds_load_b64 per lane:
//   [ a11 a12 a13 0 | a21 a22 a23 0 | r g b 0 | 0 0 0 0 ]
// lanes 0-15 read {a11,a12} (K0,K1); lanes 16-31 read {a13,0} (K2,K3).
// Padded b>=B_ records are all-zero so padded patches contribute exactly 0
// (their colours are 0). Channel 4 * orders is dead code in the reference.
// ---------------------------------------------------------------------------
__global__ void collage_prep_kernel(const float* __restrict__ tr,
                                    const float* __restrict__ rot,
                                    const float* __restrict__ sc,
                                    const float* __restrict__ sq,
                                    const float* __restrict__ sh,
                                    const float* __restrict__ re,
                                    const float* __restrict__ gr,
                                    const float* __restrict__ bl,
                                    float* __restrict__ ws) {
  int t = blockIdx.x * blockDim.x + threadIdx.x;
  if (t >= S_ * WSB_) return;
  int s = t / WSB_;
  int b = t % WSB_;
  float* o = ws + (size_t)t * REC_;
#pragma unroll
  for (int k = 0; k < REC_; ++k) o[k] = 0.f;
  if (b < B_) {
    int i = s * B_ + b;
    float r = rot[i];
    float cs = cosf(r), sn = sinf(r);
    float scv = sc[i];
    float ssv = scv * sq[i];
    float shv = sh[i];
    float tx = tr[2 * i + 0], ty = tr[2 * i + 1];
    float a11 = ssv * cs;
    float a12 = ssv * (cs * shv - sn);
    float a21 = scv * sn;
    float a22 = scv * (sn * shv + cs);
    float a13 = a11 * tx + a12 * ty;
    float a23 = a21 * tx + a22 * ty;
    o[0] = a11; o[1] = a12; o[2] = a13;            // o[3] = 0 (K3)
    o[4] = a21; o[5] = a22; o[6] = a23;            // o[7] = 0 (K3)
    o[8] = re[i]; o[9] = gr[i]; o[10] = bl[i];
  }
}

// ---------------------------------------------------------------------------
// Kernel 2: affine-grid eval via V_WMMA_F32_16X16X4_F32 (16 patches x 16
// pixels per WMMA pair), then bilinear gather of ch0..3 with zero padding and
// alpha-premultiplied accumulate over all patches.
// Wave32 layout (ISA 7.12.2, 32-bit A/B/C):
//   A 16x4:  lanes 0-15 -> row M=lane,    V0=K0, V1=K1
//            lanes16-31 -> row M=lane-16, V0=K2, V1=K3
//   B 4x16:  lanes 0-15 -> col N=lane,    V0=K0, V1=K1
//            lanes16-31 -> col N=lane-16, V0=K2, V1=K3
//   C 16x16: lane<16: N=lane, VGPR r = M=r;  lane>=16: N=lane-16, M=8+r
// => gx[M=patch][N=pixel] = a11*xs + a12*ys + a13*1 + 0
// Lane L and L+16 share pixel (L&15); each accumulates 8 of 16 patches.
// ---------------------------------------------------------------------------
__global__ __launch_bounds__(256) void collage_render_kernel(
    const float* __restrict__ patches,   // [S,B,C,H,W]
    const float* __restrict__ ws,        // [S,WSB_,REC_]
    float* __restrict__ out) {           // [S,H,W,3]
  __shared__ __align__(16) float sD[WSB_ * REC_];

  const int bid = blockIdx.x;
  const int s = bid / (TILES_X * TILES_Y);
  const int rem = bid % (TILES_X * TILES_Y);
  const int tileX = rem % TILES_X;
  const int tileY = rem / TILES_X;

  // Stage thetas + colours for this population into LDS (once per block).
  {
    const float* wsrc = ws + (size_t)s * WSB_ * REC_;
    for (int i = threadIdx.x; i < WSB_ * REC_; i += 256) sD[i] = wsrc[i];
  }
  __syncthreads();

  const int wave = threadIdx.x >> 5;      // 0..7 -> y row within tile
  const int lane = threadIdx.x & 31;
  const int lane15 = lane & 15;
  const bool hi = lane >= 16;
  const int ksel = hi ? 2 : 0;            // float offset selecting K01 vs K23

  const int px = tileX * 16 + lane15;
  const int py = tileY * 8 + wave;

  const float xs = -1.f + (2.f / (W_ - 1)) * (float)px;
  const float ys = -1.f + (2.f / (H_ - 1)) * (float)py;

  // B matrix (pixel basis [xs, ys, 1, 0]) — invariant across patch chunks.
  v2f bmat;
  bmat.x = hi ? 1.f : xs;
  bmat.y = hi ? 0.f : ys;

  float accR = 0.f, accG = 0.f, accB = 0.f;
  const float* Ps = patches + (size_t)s * B_ * C_ * HW_;

  for (int b0 = 0; b0 < WSB_; b0 += 16) {
    // A fragments: one unconditional ds_load_b64 per lane per matrix.
    const float* th = sD + (size_t)(b0 + lane15) * REC_;
    const v2f aGX = *(const v2f*)(th + ksel);        // {a11,a12} | {a13,0}
    const v2f aGY = *(const v2f*)(th + 4 + ksel);    // {a21,a22} | {a23,0}

    v8f cgx = {};
    v8f cgy = {};
    // (neg_a, A, neg_b, B, c_mod, C, reuse_a, reuse_b) — full-EXEC here.
    cgx = __builtin_amdgcn_wmma_f32_16x16x4_f32(false, aGX, false, bmat,
                                                (short)0, cgx, false, false);
    cgy = __builtin_amdgcn_wmma_f32_16x16x4_f32(false, aGY, false, bmat,
                                                (short)0, cgy, false, false);

#pragma unroll
    for (int vg = 0; vg < 8; ++vg) {
      const int b = b0 + vg + (hi ? 8 : 0);
      if (b >= B_) continue;            // divergence only in the last chunk

      const float gx = cgx[vg];
      const float gy = cgy[vg];
      // align_corners=True mapping to pixel space
      const float ix = (gx + 1.f) * (0.5f * (W_ - 1));
      const float iy = (gy + 1.f) * (0.5f * (H_ - 1));
      const float ix0f = floorf(ix);
      const float iy0f = floorf(iy);
      const float wx = ix - ix0f;
      const float wy = iy - iy0f;
      const int ix0 = (int)ix0f;
      const int iy0 = (int)iy0f;

      const float* Pb = Ps + (size_t)b * C_ * HW_;
      float a0 = 0.f, a1 = 0.f, a2 = 0.f, a3 = 0.f;

      auto corner = [&](int yy, int xx, float w) {
        if (w > 0.f && (unsigned)yy < (unsigned)H_ && (unsigned)xx < (unsigned)W_) {
          const int off = yy * W_ + xx;
          a0 = fmaf(Pb[0 * HW_ + off], w, a0);
          a1 = fmaf(Pb[1 * HW_ + off], w, a1);
          a2 = fmaf(Pb[2 * HW_ + off], w, a2);
          a3 = fmaf(Pb[3 * HW_ + off], w, a3);
        }
      };
      corner(iy0,     ix0,     (1.f - wx) * (1.f - wy));
      corner(iy0,     ix0 + 1, wx * (1.f - wy));
      corner(iy0 + 1, ix0,     (1.f - wx) * wy);
      corner(iy0 + 1, ix0 + 1, wx * wy);

      const float* cl = sD + (size_t)b * REC_ + 8;
      const float alpha = a3;           // colour for channel 3 is 1.0
      accR = fmaf(a0 * cl[0], alpha, accR);
      accG = fmaf(a1 * cl[1], alpha, accG);
      accB = fmaf(a2 * cl[2], alpha, accB);
    }
  }

  // Lane L and L+16 own the same pixel — fold halves, lanes 0-15 store.
  accR += __shfl_xor(accR, 16);
  accG += __shfl_xor(accG, 16);
  accB += __shfl_xor(accB, 16);

  if (!hi) {
    float* o = out + ((size_t)(s * H_ + py) * W_ + px) * 3;
    o[0] = fminf(fmaxf(accR, 0.f), 1.f);
    o[1] = fminf(fmaxf(accG, 0.f), 1.f);
    o[2] = fminf(fmaxf(accB, 0.f), 1.f);
  }
}

// ---------------------------------------------------------------------------
extern "C" void kernel_launch(void* const* d_in, const int* in_sizes, int n_in,
                              void* d_out, int out_size, void* d_ws, size_t ws_size,
                              hipStream_t stream) {
  const float* patches     = (const float*)d_in[0];
  const float* translation = (const float*)d_in[1];
  const float* rotation    = (const float*)d_in[2];
  const float* scale       = (const float*)d_in[3];
  const float* squeeze     = (const float*)d_in[4];
  const float* shear       = (const float*)d_in[5];
  const float* reds        = (const float*)d_in[6];
  const float* greens      = (const float*)d_in[7];
  const float* blues       = (const float*)d_in[8];
  // d_in[9] = orders: dead code in the reference (channel 4 never reaches rgb)
  float* ws  = (float*)d_ws;
  float* out = (float*)d_out;

  collage_prep_kernel<<<(S_ * WSB_ + 255) / 256, 256, 0, stream>>>(
      translation, rotation, scale, squeeze, shear, reds, greens, blues, ws);

  collage_render_kernel<<<S_ * TILES_X * TILES_Y, 256, 0, stream>>>(
      patches, ws, out);
}